// GraphConv_73469710565561
// MI455X (gfx1250) — compile-verified
//
#include <hip/hip_runtime.h>

typedef __attribute__((ext_vector_type(2))) float v2f;
typedef __attribute__((ext_vector_type(8))) float v8f;

#define IN_DIM 256
#define OUT_DIM 64
#define NEG_SLOPE 0.2f
#define L2_EPS 1e-12f

// ---------------------------------------------------------------- init
__global__ void k_init(float* __restrict__ out, float* __restrict__ e_max,
                       float* __restrict__ denom, int N) {
  int i = blockIdx.x * blockDim.x + threadIdx.x;
  if (i < N * OUT_DIM) out[i] = 0.f;
  if (i < N) { e_max[i] = -__builtin_inff(); denom[i] = 0.f; }
}

// ---------------------------------------------------------------- GEMM h = x @ W  (fp32 WMMA 16x16x4)
// block = 256 threads = 8 waves; waves 0-3 own rows 0-15 of the 32-row tile
// (one 16x16 N-tile each), waves 4-7 own rows 16-31. All of W (64 KB) and the
// 32x256 A-tile (32 KB) are staged in LDS (96 KB/workgroup; CDNA5 WGP has
// 320 KB), so the K-loop issues only ds_load + v_wmma.
__global__ __launch_bounds__(256) void k_gemm(const float* __restrict__ x,
                                              const float* __restrict__ W,
                                              float* __restrict__ h) {
  extern __shared__ float smem[];
  float* sA = smem;                 // 32*256 floats (32 KB)
  float* sW = smem + 32 * IN_DIM;   // 256*64 floats (64 KB)

  const int t = threadIdx.x;
  const size_t m0 = (size_t)blockIdx.x * 32;

  // Stage A-tile: 32 contiguous rows of x = 8192 floats = 2048 float4.
  const float4* xin = (const float4*)(x + m0 * IN_DIM);
  float4* sA4 = (float4*)sA;
#pragma unroll
  for (int i = 0; i < 8; ++i) sA4[t + i * 256] = xin[t + i * 256];

  // Stage W: 16384 floats = 4096 float4 (hits L2 after the first block).
  const float4* win = (const float4*)W;
  float4* sW4 = (float4*)sW;
#pragma unroll
  for (int i = 0; i < 16; ++i) sW4[t + i * 256] = win[t + i * 256];
  __syncthreads();

  const int lane = t & 31;
  const int wave = t >> 5;
  const int m    = lane & 15;          // row (A) / col (B,C) within tile
  const int kh   = (lane >> 4) << 1;   // lanes 0-15 -> K0,K1; lanes 16-31 -> K2,K3
  const int n    = ((wave & 3) << 4) + m;   // output column (0..63)
  const int rowoff = (wave >> 2) * 16;      // which 16-row half of the 32-row tile

  const float* sArow = sA + (rowoff + m) * IN_DIM;

  v8f c = {};
#pragma unroll 4
  for (int k0 = 0; k0 < IN_DIM; k0 += 4) {
    // A fragment: lane holds A[m][k0+kh], A[m][k0+kh+1]  (ds_load_b64)
    const float2 a2 = *(const float2*)(sArow + k0 + kh);
    v2f a = {a2.x, a2.y};
    // B fragment: lane holds W[k0+kh][n], W[k0+kh+1][n] from LDS
    v2f b = {sW[(k0 + kh) * OUT_DIM + n], sW[(k0 + kh + 1) * OUT_DIM + n]};
    // 8 args: (neg_a, A, neg_b, B, c_mod, C, reuse_a, reuse_b)
    c = __builtin_amdgcn_wmma_f32_16x16x4_f32(false, a, false, b, (short)0, c,
                                              false, false);
  }

  // C/D layout: VGPR r -> rows {r, r+8}; lanes 0-15 rows 0-7, lanes 16-31 rows 8-15.
  const int rbase = (lane >> 4) * 8;
  float* hp = h + (m0 + rowoff + rbase) * OUT_DIM + n;
#pragma unroll
  for (int r = 0; r < 8; ++r) hp[r * OUT_DIM] = c[r];
}

// ---------------------------------------------------------------- per-node scores (wave per node)
__global__ __launch_bounds__(256) void k_scores(const float* __restrict__ h,
                                                const float* __restrict__ att_src,
                                                const float* __restrict__ att_dst,
                                                float* __restrict__ a_src,
                                                float* __restrict__ a_dst, int N) {
  const int lane = threadIdx.x & 31;
  const int node = blockIdx.x * 8 + (threadIdx.x >> 5);
  if (node >= N) return;
  float2 hv = *(const float2*)(h + (size_t)node * OUT_DIM + lane * 2);
  float2 as = *(const float2*)(att_src + lane * 2);
  float2 ad = *(const float2*)(att_dst + lane * 2);
  float s = hv.x * as.x + hv.y * as.y;
  float d = hv.x * ad.x + hv.y * ad.y;
#pragma unroll
  for (int o = 16; o > 0; o >>= 1) {
    s += __shfl_xor(s, o, 32);
    d += __shfl_xor(d, o, 32);
  }
  if (lane == 0) { a_src[node] = s; a_dst[node] = d; }
}

// float atomic max via signed/unsigned int ordering trick (bit-exact for floats)
__device__ __forceinline__ void atomicMaxF(float* a, float v) {
  if (v >= 0.f) atomicMax((int*)a, __float_as_int(v));
  else          atomicMin((unsigned int*)a, __float_as_uint(v));
}

__device__ __forceinline__ void edge_sd(const long long* ei, long long E,
                                        long long t, long long& s, long long& d) {
  if (t < E) { s = ei[t]; d = ei[E + t]; }
  else       { s = d = t - E; }  // self loops appended
}

// ---------------------------------------------------------------- segment max
__global__ void k_edge_max(const long long* __restrict__ ei,
                           const float* __restrict__ a_src,
                           const float* __restrict__ a_dst,
                           float* __restrict__ e_max, long long E, long long T) {
  long long t = (long long)blockIdx.x * blockDim.x + threadIdx.x;
  if (t >= T) return;
  long long s, d;
  edge_sd(ei, E, t, s, d);
  float e = a_src[s] + a_dst[d];
  e = e > 0.f ? e : NEG_SLOPE * e;
  atomicMaxF(&e_max[d], e);
}

// ---------------------------------------------------------------- exp + segment sum
__global__ void k_edge_exp(const long long* __restrict__ ei,
                           const float* __restrict__ a_src,
                           const float* __restrict__ a_dst,
                           const float* __restrict__ e_max,
                           float* __restrict__ e_exp, float* __restrict__ denom,
                           long long E, long long T) {
  long long t = (long long)blockIdx.x * blockDim.x + threadIdx.x;
  if (t >= T) return;
  long long s, d;
  edge_sd(ei, E, t, s, d);
  float e = a_src[s] + a_dst[d];
  e = e > 0.f ? e : NEG_SLOPE * e;
  float v = expf(e - e_max[d]);
  e_exp[t] = v;
  atomicAdd(&denom[d], v);
}

// ---------------------------------------------------------------- scatter-aggregate (wave per edge)
__global__ __launch_bounds__(256) void k_aggregate(
    const long long* __restrict__ ei, const float* __restrict__ h,
    const float* __restrict__ e_exp, const float* __restrict__ denom,
    float* __restrict__ out, long long E, long long T) {
  const int lane = threadIdx.x & 31;
  long long t = (long long)blockIdx.x * 8 + (threadIdx.x >> 5);
  if (t >= T) return;
  long long s, d;
  edge_sd(ei, E, t, s, d);
  // warm the h[src] row while alpha is computed (global_prefetch_b8)
  __builtin_prefetch(h + (size_t)s * OUT_DIM + lane * 2, 0, 3);
  float alpha = 0.f;
  if (lane == 0) alpha = e_exp[t] / denom[d];
  alpha = __shfl(alpha, 0, 32);
  float2 hv = *(const float2*)(h + (size_t)s * OUT_DIM + lane * 2);
  float* op = out + (size_t)d * OUT_DIM + lane * 2;
  atomicAdd(op + 0, alpha * hv.x);
  atomicAdd(op + 1, alpha * hv.y);
}

// ---------------------------------------------------------------- bias + row L2 normalize (in place)
__global__ __launch_bounds__(256) void k_finalize(float* __restrict__ out,
                                                  const float* __restrict__ bias,
                                                  int N) {
  const int lane = threadIdx.x & 31;
  const int node = blockIdx.x * 8 + (threadIdx.x >> 5);
  if (node >= N) return;
  float* op = out + (size_t)node * OUT_DIM + lane * 2;
  float2 v = *(const float2*)op;
  v.x += bias[lane * 2];
  v.y += bias[lane * 2 + 1];
  float ss = v.x * v.x + v.y * v.y;
#pragma unroll
  for (int o = 16; o > 0; o >>= 1) ss += __shfl_xor(ss, o, 32);
  float scale = 1.f / fmaxf(sqrtf(ss), L2_EPS);
  op[0] = v.x * scale;
  op[1] = v.y * scale;
}

// ---------------------------------------------------------------- launcher
extern "C" void kernel_launch(void* const* d_in, const int* in_sizes, int n_in,
                              void* d_out, int out_size, void* d_ws, size_t ws_size,
                              hipStream_t stream) {
  const float*     x       = (const float*)d_in[0];
  const long long* ei      = (const long long*)d_in[1];
  const float*     W       = (const float*)d_in[2];
  const float*     att_src = (const float*)d_in[3];
  const float*     att_dst = (const float*)d_in[4];
  const float*     bias    = (const float*)d_in[5];
  float*           out     = (float*)d_out;

  const long long N = (long long)in_sizes[0] / IN_DIM;  // 100000 (multiple of 32)
  const long long E = (long long)in_sizes[1] / 2;       // 1600000
  const long long T = E + N;                            // edges + self loops

  // workspace layout (floats)
  float* ws    = (float*)d_ws;
  float* h     = ws;                    // N*64
  float* a_src = h + (size_t)N * OUT_DIM;
  float* a_dst = a_src + N;
  float* e_max = a_dst + N;
  float* denom = e_max + N;
  float* e_exp = denom + N;             // T

  const size_t gemm_lds = (size_t)(32 * IN_DIM + IN_DIM * OUT_DIM) * sizeof(float); // 96 KB

  k_init<<<(unsigned)((N * OUT_DIM + 255) / 256), 256, 0, stream>>>(out, e_max, denom, (int)N);
  k_gemm<<<(unsigned)(N / 32), 256, gemm_lds, stream>>>(x, W, h);  // N is an exact multiple of 32
  k_scores<<<(unsigned)((N + 7) / 8), 256, 0, stream>>>(h, att_src, att_dst, a_src, a_dst, (int)N);
  k_edge_max<<<(unsigned)((T + 255) / 256), 256, 0, stream>>>(ei, a_src, a_dst, e_max, E, T);
  k_edge_exp<<<(unsigned)((T + 255) / 256), 256, 0, stream>>>(ei, a_src, a_dst, e_max, e_exp, denom, E, T);
  k_aggregate<<<(unsigned)((T + 7) / 8), 256, 0, stream>>>(ei, h, e_exp, denom, out, E, T);
  k_finalize<<<(unsigned)((N + 7) / 8), 256, 0, stream>>>(out, bias, (int)N);
}